// Classifier_6760278524368
// MI455X (gfx1250) — compile-verified
//
#include <hip/hip_runtime.h>
#include <hip/hip_bf16.h>

typedef float v2f __attribute__((ext_vector_type(2)));
typedef float v8f __attribute__((ext_vector_type(8)));

#define DH 64   // feature width of both GCN layers (D == H == 64)

// ---------------------------------------------------------------------------
// grid-stride fp32 zero
// ---------------------------------------------------------------------------
__global__ void zero_f32_kernel(float* __restrict__ p, long n) {
    long i = (long)blockIdx.x * blockDim.x + threadIdx.x;
    long stride = (long)gridDim.x * blockDim.x;
    for (; i < n; i += stride) p[i] = 0.0f;
}

// ---------------------------------------------------------------------------
// degree accumulation: out_deg[src[e]] += 1, in_deg[dst[e]] += 1
// ---------------------------------------------------------------------------
__global__ void degree_kernel(const int* __restrict__ src, const int* __restrict__ dst,
                              float* __restrict__ deg_out, float* __restrict__ deg_in, int E) {
    int e = blockIdx.x * blockDim.x + threadIdx.x;
    if (e < E) {
        atomicAdd(&deg_out[src[e]], 1.0f);
        atomicAdd(&deg_in[dst[e]], 1.0f);
    }
}

// ---------------------------------------------------------------------------
// deg -> rsqrt(max(deg,1)) in place; also per-graph node counts
// ---------------------------------------------------------------------------
__global__ void norm_count_kernel(float* __restrict__ nsrc, float* __restrict__ ndst,
                                  const int* __restrict__ gid, float* __restrict__ cnt, int N) {
    int i = blockIdx.x * blockDim.x + threadIdx.x;
    if (i < N) {
        nsrc[i] = rsqrtf(fmaxf(nsrc[i], 1.0f));
        ndst[i] = rsqrtf(fmaxf(ndst[i], 1.0f));
        atomicAdd(&cnt[gid[i]], 1.0f);
    }
}

// ---------------------------------------------------------------------------
// edge aggregation: agg[dst] += x[src] * norm_src[src]
// 16 threads per edge, float4 gather (L2-resident), 4 f32 atomics scatter
// ---------------------------------------------------------------------------
__global__ void aggregate_kernel(const float* __restrict__ x, const float* __restrict__ norm_src,
                                 const int* __restrict__ src, const int* __restrict__ dst,
                                 float* __restrict__ agg, int E) {
    long t = (long)blockIdx.x * blockDim.x + threadIdx.x;
    int e = (int)(t >> 4);        // edge id
    int c = (int)(t & 15) << 2;   // feature chunk base (0..60 step 4)
    if (e >= E) return;
    int s = src[e];
    int d = dst[e];
    float ns = norm_src[s];
    const float4 v = *(const float4*)(x + (size_t)s * DH + c);
    float* o = agg + (size_t)d * DH + c;
    atomicAdd(o + 0, v.x * ns);
    atomicAdd(o + 1, v.y * ns);
    atomicAdd(o + 2, v.z * ns);
    atomicAdd(o + 3, v.w * ns);
}

// ---------------------------------------------------------------------------
// fused: out = relu((agg * norm_dst[row]) @ W + b), via V_WMMA_F32_16X16X4_F32.
// One wave owns a 16-row x 64-col tile: 4 accumulators, 16 k-steps of K=4.
// POOL=true: instead of storing rows, atomically pool into pooled[graph,64].
// ---------------------------------------------------------------------------
template <bool POOL>
__global__ void gcn_gemm_relu_kernel(const float* __restrict__ agg,
                                     const float* __restrict__ norm_dst,
                                     const float* __restrict__ W,    // [64,64] row-major (k, col)
                                     const float* __restrict__ bias, // [64]
                                     float* __restrict__ out,        // POOL ? pooled[G*64] : h_out[N*64]
                                     const int* __restrict__ gid,    // graph ids (POOL only)
                                     int nrows) {
    __shared__ float Wsh[DH * DH];
    __shared__ float Bsh[DH];
    for (int i = threadIdx.x; i < DH * DH; i += blockDim.x) Wsh[i] = W[i];
    if (threadIdx.x < DH) Bsh[threadIdx.x] = bias[threadIdx.x];
    __syncthreads();

    const int lane  = threadIdx.x & 31;   // wave32
    const int wave  = threadIdx.x >> 5;   // 8 waves / block
    const int m     = lane & 15;          // row-in-tile (A) / col-in-tile (B,D)
    const int khalf = lane >> 4;          // 0: K pair {0,1}; 1: K pair {2,3}
    const int r0    = blockIdx.x * (8 * 16) + wave * 16;
    if (r0 >= nrows) return;              // wave-uniform guard (after syncthreads)

    const int   row  = (r0 + m < nrows) ? (r0 + m) : (nrows - 1);
    const float nd   = norm_dst[row];
    const float* arow = agg + (size_t)row * DH;

    v8f acc0 = {}, acc1 = {}, acc2 = {}, acc3 = {};

#pragma unroll
    for (int ks = 0; ks < 16; ++ks) {
        const int ka = ks * 4 + 2 * khalf;     // this lane-half's K pair
        // A fragment: element (m, ka), (m, ka+1), scaled by norm_dst
        v2f a;
        {
            const float2 av = *(const float2*)(arow + ka);
            a.x = av.x * nd;
            a.y = av.y * nd;
        }
        // B fragments: vgpr j holds row K = ka + j, col = 16*n + m
        const float* w0 = Wsh + ka * DH;
        const float* w1 = w0 + DH;
        v2f b0, b1, b2, b3;
        b0.x = w0[m];       b0.y = w1[m];
        b1.x = w0[16 + m];  b1.y = w1[16 + m];
        b2.x = w0[32 + m];  b2.y = w1[32 + m];
        b3.x = w0[48 + m];  b3.y = w1[48 + m];

        acc0 = __builtin_amdgcn_wmma_f32_16x16x4_f32(false, a, false, b0, (short)0, acc0, false, false);
        acc1 = __builtin_amdgcn_wmma_f32_16x16x4_f32(false, a, false, b1, (short)0, acc1, false, false);
        acc2 = __builtin_amdgcn_wmma_f32_16x16x4_f32(false, a, false, b2, (short)0, acc2, false, false);
        acc3 = __builtin_amdgcn_wmma_f32_16x16x4_f32(false, a, false, b3, (short)0, acc3, false, false);
    }

    // Epilogue. D layout: vgpr j -> row (j + khalf*8), col = lane&15 within tile.
#pragma unroll
    for (int j = 0; j < 8; ++j) {
        const int orow = r0 + j + khalf * 8;
        if (orow >= nrows) continue;
        const float v0 = fmaxf(acc0[j] + Bsh[m],      0.0f);
        const float v1 = fmaxf(acc1[j] + Bsh[16 + m], 0.0f);
        const float v2 = fmaxf(acc2[j] + Bsh[32 + m], 0.0f);
        const float v3 = fmaxf(acc3[j] + Bsh[48 + m], 0.0f);
        if (POOL) {
            const int g = gid[orow];
            float* p = out + (size_t)g * DH;
            atomicAdd(p + m,      v0);
            atomicAdd(p + 16 + m, v1);
            atomicAdd(p + 32 + m, v2);
            atomicAdd(p + 48 + m, v3);
        } else {
            float* p = out + (size_t)orow * DH;
            p[m]      = v0;
            p[16 + m] = v1;
            p[32 + m] = v2;
            p[48 + m] = v3;
        }
    }
}

// ---------------------------------------------------------------------------
// logits[g,c] = (pooled[g,:] / max(cnt[g],1)) @ Wc[:,c] + bc[c]
// ---------------------------------------------------------------------------
__global__ void classifier_kernel(const float* __restrict__ pooled, const float* __restrict__ cnt,
                                  const float* __restrict__ Wc, const float* __restrict__ bc,
                                  float* __restrict__ out, int G, int C) {
    int t = blockIdx.x * blockDim.x + threadIdx.x;
    if (t >= G * C) return;
    int g = t / C;
    int c = t - g * C;
    float inv = 1.0f / fmaxf(cnt[g], 1.0f);
    const float* pg = pooled + (size_t)g * DH;
    float acc = bc[c];
#pragma unroll
    for (int d = 0; d < DH; ++d) acc = fmaf(pg[d] * inv, Wc[d * C + c], acc);
    out[t] = acc;
}

// ---------------------------------------------------------------------------
extern "C" void kernel_launch(void* const* d_in, const int* in_sizes, int n_in,
                              void* d_out, int out_size, void* d_ws, size_t ws_size,
                              hipStream_t stream) {
    const float* h   = (const float*)d_in[0];
    const int*   src = (const int*)d_in[1];
    const int*   dst = (const int*)d_in[2];
    const int*   gid = (const int*)d_in[3];
    const float* W1  = (const float*)d_in[4];
    const float* b1  = (const float*)d_in[5];
    const float* W2  = (const float*)d_in[6];
    const float* b2  = (const float*)d_in[7];
    const float* Wc  = (const float*)d_in[8];
    const float* bc  = (const float*)d_in[9];
    float* out = (float*)d_out;

    const int N = in_sizes[3];           // graph_ids length == #nodes
    const int E = in_sizes[1];           // #edges
    const int C = in_sizes[9];           // #classes (bc length)
    const int G = out_size / C;          // #graphs

    // workspace layout (fp32)
    float* ws       = (float*)d_ws;
    float* norm_src = ws;                                 // N   (deg_out -> rsqrt)
    float* norm_dst = norm_src + N;                       // N   (deg_in  -> rsqrt)
    float* cnt      = norm_dst + N;                       // G
    float* pooled   = cnt + G;                            // G*64
    float* aggbuf   = pooled + (size_t)G * DH;            // N*64
    float* hbuf     = aggbuf + (size_t)N * DH;            // N*64 (h1, layer-2 input)

    const long zero_n = 2L * N + G + (long)G * DH + (long)N * DH; // everything up to aggbuf end
    const int  nblk_rows = (N + 127) / 128;               // 8 waves x 16 rows per block
    const long agg_thr = (long)E * 16;

    zero_f32_kernel<<<1024, 256, 0, stream>>>(ws, zero_n);
    degree_kernel<<<(E + 255) / 256, 256, 0, stream>>>(src, dst, norm_src, norm_dst, E);
    norm_count_kernel<<<(N + 255) / 256, 256, 0, stream>>>(norm_src, norm_dst, gid, cnt, N);

    // layer 1
    aggregate_kernel<<<(int)((agg_thr + 255) / 256), 256, 0, stream>>>(h, norm_src, src, dst, aggbuf, E);
    gcn_gemm_relu_kernel<false><<<nblk_rows, 256, 0, stream>>>(aggbuf, norm_dst, W1, b1, hbuf, nullptr, N);

    // layer 2 (pool fused into epilogue; h2 never materialized)
    zero_f32_kernel<<<1024, 256, 0, stream>>>(aggbuf, (long)N * DH);
    aggregate_kernel<<<(int)((agg_thr + 255) / 256), 256, 0, stream>>>(hbuf, norm_src, src, dst, aggbuf, E);
    gcn_gemm_relu_kernel<true><<<nblk_rows, 256, 0, stream>>>(aggbuf, norm_dst, W2, b2, pooled, gid, N);

    // readout
    classifier_kernel<<<(G * C + 255) / 256, 256, 0, stream>>>(pooled, cnt, Wc, bc, out, G, C);
}